// _ProposalLayer_23931557773521
// MI455X (gfx1250) — compile-verified
//
#include <hip/hip_runtime.h>
#include <stdint.h>

// ---------------- problem constants (from reference) ----------------
#define A_NUM 12
#define T_DIM 8
#define H_DIM 64
#define W_DIM 64
#define HW_SZ   (H_DIM * W_DIM)      // 4096
#define THW_SZ  (T_DIM * HW_SZ)      // 32768
#define NFG     (A_NUM * THW_SZ)     // 393216 fg scores per batch
#define TOPK    100
#define NBINS   4096                 // key >> 20 : sign + 8 exp + 3 mantissa bits
#define CAND_CAP 2048
#define CHUNK   4096                 // floats per block in streaming passes
#define BPB     (NFG / CHUNK)        // 96 blocks per batch
#define FEAT_STRIDE_F 16.0f

// async staging pipeline geometry (wave32: 8 waves per 256-thread block)
#define WAVES_PER_BLK 8
#define WAVE_FLOATS  (CHUNK / WAVES_PER_BLK)   // 512 floats per wave per block
#define STAGES       4
#define STAGE_FLOATS (WAVE_FLOATS / STAGES)    // 128 floats per wave per stage (lane*4)

// Precomputed _generate_anchors() output, cols {x1,y1,x2,y2} (z1=0, z2=15 constant).
__constant__ float c_anc[A_NUM][4] = {
    {-38.f, -16.f,  53.f,  31.f},   // ratio 0.5, scale 4
    {-84.f, -40.f,  99.f,  55.f},   // ratio 0.5, scale 8
    {-176.f,-88.f, 191.f, 103.f},   // ratio 0.5, scale 16
    {-360.f,-184.f,375.f, 199.f},   // ratio 0.5, scale 32
    {-24.f, -24.f,  39.f,  39.f},   // ratio 1.0, scale 4
    {-56.f, -56.f,  71.f,  71.f},   // ratio 1.0, scale 8
    {-120.f,-120.f,135.f, 135.f},   // ratio 1.0, scale 16
    {-248.f,-248.f,263.f, 263.f},   // ratio 1.0, scale 32
    {-14.f, -36.f,  29.f,  51.f},   // ratio 2.0, scale 4
    {-36.f, -80.f,  51.f,  95.f},   // ratio 2.0, scale 8
    {-80.f, -168.f, 95.f, 183.f},   // ratio 2.0, scale 16
    {-168.f,-344.f,183.f, 359.f},   // ratio 2.0, scale 32
};

// Monotone float->uint key: preserves ordering, invertible.
__device__ __forceinline__ unsigned key_of(float s) {
    unsigned u = __float_as_uint(s);
    return (u & 0x80000000u) ? ~u : (u | 0x80000000u);
}
__device__ __forceinline__ float float_of_key(unsigned k) {
    unsigned u = (k & 0x80000000u) ? (k & 0x7FFFFFFFu) : ~k;
    return __uint_as_float(u);
}

// CDNA5 async global->LDS B128: lds byte-offset in a VGPR, 32-bit global offset
// VGPR + 64-bit SGPR base (GVS mode). Tracked by ASYNCcnt.
__device__ __forceinline__ void async_load_b128(unsigned lds_off, unsigned g_off,
                                                const float* base) {
    asm volatile("global_load_async_to_lds_b128 %0, %1, %2"
                 :: "v"(lds_off), "v"(g_off), "s"(base) : "memory");
}

// ---------------- pass 0: zero workspace ----------------
__global__ void k_init(unsigned* __restrict__ hist, unsigned* __restrict__ cnt,
                       int nhist, int B) {
    int i = blockIdx.x * blockDim.x + threadIdx.x;
    if (i < nhist) hist[i] = 0u;
    if (i < B)     cnt[i]  = 0u;
}

// ---------------- pass 1: per-batch 4096-bin histogram of score keys ----------------
// Streams fg scores through LDS with the gfx1250 async copy engine: each wave
// issues 4 in-flight B128 copies into its private LDS slice, then drains them
// with staggered s_wait_asynccnt (async loads complete in order; wave-private
// slices need no workgroup barrier).
__global__ void k_hist(const float* __restrict__ scores, unsigned* __restrict__ hist) {
    __shared__ unsigned sh[NBINS];
    __shared__ float    sbuf[WAVES_PER_BLK * WAVE_FLOATS];   // 16 KB staging
    const int b = blockIdx.x / BPB;
    const int c = blockIdx.x % BPB;
    for (int i = threadIdx.x; i < NBINS; i += blockDim.x) sh[i] = 0u;
    __syncthreads();

    const unsigned lane = threadIdx.x & 31u;
    const unsigned wave = threadIdx.x >> 5;
    const float* base = scores + ((size_t)b * 2 * A_NUM + A_NUM) * THW_SZ
                               + (size_t)c * CHUNK + (size_t)wave * WAVE_FLOATS;
    float* wbuf = &sbuf[wave * WAVE_FLOATS];
    const unsigned lds_base = (unsigned)(uintptr_t)wbuf;  // LDS aperture: low 32 bits = offset

    // gfx1250 speculative prefetch of the next block's chunk (global_prefetch_b8)
    __builtin_prefetch((const char*)(base + CHUNK) + lane * 64, 0, 0);

#pragma unroll
    for (int s = 0; s < STAGES; ++s) {
        const unsigned goff = (unsigned)((s * STAGE_FLOATS + lane * 4) * sizeof(float));
        async_load_b128(lds_base + goff, goff, base);
    }
    {
        asm volatile("s_wait_asynccnt 3" ::: "memory");
        const float4 v = *(const float4*)&wbuf[0 * STAGE_FLOATS + lane * 4];
        atomicAdd(&sh[key_of(v.x) >> 20], 1u); atomicAdd(&sh[key_of(v.y) >> 20], 1u);
        atomicAdd(&sh[key_of(v.z) >> 20], 1u); atomicAdd(&sh[key_of(v.w) >> 20], 1u);
    }
    {
        asm volatile("s_wait_asynccnt 2" ::: "memory");
        const float4 v = *(const float4*)&wbuf[1 * STAGE_FLOATS + lane * 4];
        atomicAdd(&sh[key_of(v.x) >> 20], 1u); atomicAdd(&sh[key_of(v.y) >> 20], 1u);
        atomicAdd(&sh[key_of(v.z) >> 20], 1u); atomicAdd(&sh[key_of(v.w) >> 20], 1u);
    }
    {
        asm volatile("s_wait_asynccnt 1" ::: "memory");
        const float4 v = *(const float4*)&wbuf[2 * STAGE_FLOATS + lane * 4];
        atomicAdd(&sh[key_of(v.x) >> 20], 1u); atomicAdd(&sh[key_of(v.y) >> 20], 1u);
        atomicAdd(&sh[key_of(v.z) >> 20], 1u); atomicAdd(&sh[key_of(v.w) >> 20], 1u);
    }
    {
        asm volatile("s_wait_asynccnt 0" ::: "memory");
        const float4 v = *(const float4*)&wbuf[3 * STAGE_FLOATS + lane * 4];
        atomicAdd(&sh[key_of(v.x) >> 20], 1u); atomicAdd(&sh[key_of(v.y) >> 20], 1u);
        atomicAdd(&sh[key_of(v.z) >> 20], 1u); atomicAdd(&sh[key_of(v.w) >> 20], 1u);
    }
    __syncthreads();

    unsigned* gh = hist + (size_t)b * NBINS;
    for (int i = threadIdx.x; i < NBINS; i += blockDim.x) {
        unsigned v = sh[i];
        if (v) atomicAdd(&gh[i], v);
    }
}

// ---------------- pass 2: find cutoff bin per batch ----------------
__global__ void k_cutoff(const unsigned* __restrict__ hist, unsigned* __restrict__ meta, int B) {
    int b = threadIdx.x;
    if (b >= B) return;
    const unsigned* h = hist + (size_t)b * NBINS;
    unsigned cum = 0; int cut = 0;
    for (int bin = NBINS - 1; bin >= 0; --bin) {
        unsigned c = h[bin];
        if (cum + c >= TOPK) { cut = bin; break; }
        cum += c;
    }
    meta[b] = (unsigned)cut;
}

// ---------------- pass 3: collect candidates (key, flat-proposal-index) ----------------
// Same async-staged streaming as k_hist (this pass largely hits L2 / 192 MB).
__global__ void k_collect(const float* __restrict__ scores, const unsigned* __restrict__ meta,
                          unsigned* __restrict__ cnt, uint2* __restrict__ cand) {
    __shared__ float sbuf[WAVES_PER_BLK * WAVE_FLOATS];   // 16 KB staging
    const int b = blockIdx.x / BPB;
    const int c = blockIdx.x % BPB;
    const unsigned cut = meta[b];

    const unsigned lane = threadIdx.x & 31u;
    const unsigned wave = threadIdx.x >> 5;
    const float* base = scores + ((size_t)b * 2 * A_NUM + A_NUM) * THW_SZ
                               + (size_t)c * CHUNK + (size_t)wave * WAVE_FLOATS;
    float* wbuf = &sbuf[wave * WAVE_FLOATS];
    const unsigned lds_base = (unsigned)(uintptr_t)wbuf;
    uint2* cb = cand + (size_t)b * CAND_CAP;
    const unsigned ebase = (unsigned)c * CHUNK + wave * WAVE_FLOATS;

#pragma unroll
    for (int s = 0; s < STAGES; ++s) {
        const unsigned goff = (unsigned)((s * STAGE_FLOATS + lane * 4) * sizeof(float));
        async_load_b128(lds_base + goff, goff, base);
    }
#pragma unroll
    for (int s = 0; s < STAGES; ++s) {
        switch (s) {   // immediate operand for the counter wait
            case 0: asm volatile("s_wait_asynccnt 3" ::: "memory"); break;
            case 1: asm volatile("s_wait_asynccnt 2" ::: "memory"); break;
            case 2: asm volatile("s_wait_asynccnt 1" ::: "memory"); break;
            default: asm volatile("s_wait_asynccnt 0" ::: "memory"); break;
        }
        const float4 v = *(const float4*)&wbuf[s * STAGE_FLOATS + lane * 4];
        const float vv[4] = {v.x, v.y, v.z, v.w};
        const unsigned e0 = ebase + (unsigned)(s * STAGE_FLOATS + lane * 4);
#pragma unroll
        for (int q = 0; q < 4; ++q) {
            unsigned k = key_of(vv[q]);
            if ((k >> 20) >= cut) {
                unsigned e = e0 + q;                            // e = a*THW + t*HW + h*W + w
                unsigned n = (e % THW_SZ) * A_NUM + e / THW_SZ; // ref ordering: (t,h,w)*A + a
                unsigned pos = atomicAdd(&cnt[b], 1u);
                if (pos < CAND_CAP) cb[pos] = make_uint2(k, n);
            }
        }
    }
}

// ---------------- pass 4: exact top-100 + decode winners ----------------
__global__ void k_final(const uint2* __restrict__ cand, const unsigned* __restrict__ cnt,
                        const float* __restrict__ bbox, const float* __restrict__ iminfo,
                        float* __restrict__ out) {
    __shared__ unsigned skey[CAND_CAP];
    __shared__ unsigned sidx[CAND_CAP];
    __shared__ unsigned rkey[256], ridx[256], rpos[256];
    __shared__ unsigned wkey[TOPK], widx[TOPK];

    const int b = blockIdx.x;
    const unsigned m = min(cnt[b], (unsigned)CAND_CAP);
    const uint2* cb = cand + (size_t)b * CAND_CAP;

    for (int j = threadIdx.x; j < CAND_CAP; j += 256) {
        if (j < (int)m) { uint2 v = cb[j]; skey[j] = v.x; sidx[j] = v.y; }
        else            { skey[j] = 0u; sidx[j] = 0xFFFFFFFFu; }
    }
    __syncthreads();

    // 100 rounds of argmax with deterministic tie-break (key desc, index asc)
    for (int k = 0; k < TOPK; ++k) {
        unsigned bk = 0u, bi = 0xFFFFFFFFu, bp = 0u;
        for (int j = threadIdx.x; j < CAND_CAP; j += 256) {
            unsigned kk = skey[j], ii = sidx[j];
            if (kk > bk || (kk == bk && ii < bi)) { bk = kk; bi = ii; bp = (unsigned)j; }
        }
        rkey[threadIdx.x] = bk; ridx[threadIdx.x] = bi; rpos[threadIdx.x] = bp;
        __syncthreads();
        for (int s = 128; s > 0; s >>= 1) {
            if (threadIdx.x < s) {
                unsigned ok = rkey[threadIdx.x + s], oi = ridx[threadIdx.x + s];
                if (ok > rkey[threadIdx.x] ||
                    (ok == rkey[threadIdx.x] && oi < ridx[threadIdx.x])) {
                    rkey[threadIdx.x] = ok; ridx[threadIdx.x] = oi;
                    rpos[threadIdx.x] = rpos[threadIdx.x + s];
                }
            }
            __syncthreads();
        }
        if (threadIdx.x == 0) {
            wkey[k] = rkey[0]; widx[k] = ridx[0];
            skey[rpos[0]] = 0u; sidx[rpos[0]] = 0xFFFFFFFFu;  // remove winner
        }
        __syncthreads();
    }

    // decode the 100 winners in parallel (only 6 delta loads each)
    if (threadIdx.x < TOPK) {
        const int k = threadIdx.x;
        const unsigned n = widx[k];
        float* o = out + ((size_t)b * TOPK + k) * 8;
        if (n == 0xFFFFFFFFu) {                 // unreachable for this problem size; safety
            o[0] = (float)b;
            for (int q = 1; q < 8; ++q) o[q] = 0.f;
            return;
        }
        const float score = float_of_key(wkey[k]);
        const unsigned a = n % A_NUM;
        const unsigned i = n / A_NUM;           // i = t*HW + h*W + w  (score/delta position)
        // anchor-shift decode uses the reference's (h',w',t') ordering of the SAME i:
        const unsigned tp = i & 7u, wp = (i >> 3) & 63u, hp = i >> 9;
        const float sx = (float)wp * FEAT_STRIDE_F;
        const float sy = (float)hp * FEAT_STRIDE_F;
        const float sz = (float)tp;

        const float x1 = c_anc[a][0] + sx, y1 = c_anc[a][1] + sy;
        const float x2 = c_anc[a][2] + sx, y2 = c_anc[a][3] + sy;
        const float z1 = sz, z2 = 15.0f + sz;
        const float wid = x2 - x1 + 1.0f, hei = y2 - y1 + 1.0f, dur = z2 - z1 + 1.0f;
        const float cx = x1 + 0.5f * wid, cy = y1 + 0.5f * hei, ct = z1 + 0.5f * dur;

        const float* dbase = bbox + ((size_t)b * 6 * A_NUM + (size_t)a * 6) * THW_SZ + i;
        const float d0 = dbase[0];
        const float d1 = dbase[1 * THW_SZ];
        const float d2 = dbase[2 * THW_SZ];
        const float d3 = dbase[3 * THW_SZ];
        const float d4 = dbase[4 * THW_SZ];
        const float d5 = dbase[5 * THW_SZ];

        const float px = d0 * wid + cx, py = d1 * hei + cy, pt = d2 * dur + ct;
        const float pw = expf(d3) * wid, ph = expf(d4) * hei, pd = expf(d5) * dur;
        float p0 = px - 0.5f * pw, p1 = py - 0.5f * ph, p2 = pt - 0.5f * pd;
        float p3 = px + 0.5f * pw, p4 = py + 0.5f * ph, p5 = pt + 0.5f * pd;

        const float hiX = iminfo[b * 3 + 1] - 1.0f;
        const float hiY = iminfo[b * 3 + 0] - 1.0f;
        const float hiT = iminfo[b * 3 + 2] - 1.0f;
        p0 = fminf(fmaxf(p0, 0.f), hiX); p3 = fminf(fmaxf(p3, 0.f), hiX);
        p1 = fminf(fmaxf(p1, 0.f), hiY); p4 = fminf(fmaxf(p4, 0.f), hiY);
        p2 = fminf(fmaxf(p2, 0.f), hiT); p5 = fminf(fmaxf(p5, 0.f), hiT);

        o[0] = (float)b;
        o[1] = p0; o[2] = p1; o[3] = p2;
        o[4] = p3; o[5] = p4; o[6] = p5;
        o[7] = score;
    }
}

extern "C" void kernel_launch(void* const* d_in, const int* in_sizes, int n_in,
                              void* d_out, int out_size, void* d_ws, size_t ws_size,
                              hipStream_t stream) {
    const float* scores = (const float*)d_in[0];
    const float* bbox   = (const float*)d_in[1];
    const float* iminfo = (const float*)d_in[2];
    float*       out    = (float*)d_out;

    const int B = in_sizes[0] / (2 * A_NUM * THW_SZ);   // 8 for the reference shapes

    // workspace layout
    unsigned char* ws = (unsigned char*)d_ws;
    unsigned* hist = (unsigned*)ws;                       // B*NBINS u32
    size_t histBytes = (size_t)B * NBINS * sizeof(unsigned);
    unsigned* meta = (unsigned*)(ws + histBytes);         // B u32 (cutoff bin)
    unsigned* cnt  = meta + B;                            // B u32 (candidate count)
    size_t candOff = ((histBytes + 2 * (size_t)B * sizeof(unsigned) + 15) / 16) * 16;
    uint2* cand = (uint2*)(ws + candOff);                 // B*CAND_CAP uint2

    const int nhist = B * NBINS;
    k_init   <<<(nhist + 255) / 256, 256, 0, stream>>>(hist, cnt, nhist, B);
    k_hist   <<<B * BPB, 256, 0, stream>>>(scores, hist);
    k_cutoff <<<1, 32, 0, stream>>>(hist, meta, B);
    k_collect<<<B * BPB, 256, 0, stream>>>(scores, meta, cnt, cand);
    k_final  <<<B, 256, 0, stream>>>(cand, cnt, bbox, iminfo, out);
}